// DOMINANT_86045374808287
// MI455X (gfx1250) — compile-verified
//
#include <hip/hip_runtime.h>

typedef float v2f __attribute__((ext_vector_type(2)));
typedef float v8f __attribute__((ext_vector_type(8)));

// ---------------------------------------------------------------------------
// Degree / normalization
// ---------------------------------------------------------------------------
__global__ __launch_bounds__(256) void k_init_deg(float* __restrict__ deg, int n) {
    int i = blockIdx.x * blockDim.x + threadIdx.x;
    if (i < n) deg[i] = 1.0f;  // self-loop contributes 1 to every node's degree
}

__global__ __launch_bounds__(256) void k_count_deg(const long long* __restrict__ ei,
                                                   float* __restrict__ deg, int E) {
    int e = blockIdx.x * blockDim.x + threadIdx.x;
    if (e < E) {
        long long d = ei[(long long)E + e];  // dst row of edge_index
        unsafeAtomicAdd(&deg[d], 1.0f);
    }
}

__global__ __launch_bounds__(256) void k_deg_inv_sqrt(float* __restrict__ deg, int n) {
    int i = blockIdx.x * blockDim.x + threadIdx.x;
    if (i < n) {
        float d = deg[i];
        deg[i] = (d > 0.0f) ? rsqrtf(d) : 0.0f;
    }
}

// ---------------------------------------------------------------------------
// Dense GEMM H[M,Nout] = X[M,K] * W[K,Nout] using V_WMMA_F32_16X16X4_F32.
// One wave computes one 16x16 tile of H, stepping K by 4.
//
// ISA lane layouts (wave32):
//   A (16x4, M x K): lane = half*16 + lo; a[j] = A[lo][k + 2*half + j]
//   B (4x16, K x N): b[j] = W[(k + 2*half + j)][n0 + lo]
//   C/D (16x16):     c[v] = D[m0 + v + 8*half][n0 + lo]
// ---------------------------------------------------------------------------
__global__ __launch_bounds__(256) void k_gemm_wmma_f32(const float* __restrict__ X,
                                                       const float* __restrict__ W,
                                                       float* __restrict__ H,
                                                       int M, int K, int Nout) {
    int wave  = blockIdx.x * (blockDim.x >> 5) + (threadIdx.x >> 5);
    int lane  = threadIdx.x & 31;
    int tn_n  = Nout >> 4;
    int tiles = (M >> 4) * tn_n;
    if (wave >= tiles) return;  // wave-uniform branch: EXEC stays all-ones

    int tm = wave / tn_n;
    int tn = wave - tm * tn_n;
    int m0 = tm << 4;
    int n0 = tn << 4;
    int half = lane >> 4;
    int lo   = lane & 15;

    v8f c = {0.f, 0.f, 0.f, 0.f, 0.f, 0.f, 0.f, 0.f};

    const float* arow = X + (size_t)(m0 + lo) * K + 2 * half;   // + k each step
    const float* bcol = W + (size_t)(2 * half) * Nout + n0 + lo; // + k*Nout each step

    for (int k = 0; k < K; k += 4) {
        v2f a = *(const v2f*)(arow + k);           // A[lo][k+2h], A[lo][k+2h+1]
        v2f b;
        const float* bp = bcol + (size_t)k * Nout;
        b.x = bp[0];                               // W[k+2h  ][n0+lo]
        b.y = bp[Nout];                            // W[k+2h+1][n0+lo]
        c = __builtin_amdgcn_wmma_f32_16x16x4_f32(false, a, false, b,
                                                  (short)0, c, false, false);
    }

#pragma unroll
    for (int v = 0; v < 8; ++v)
        H[(size_t)(m0 + v + 8 * half) * Nout + n0 + lo] = c[v];
}

// ---------------------------------------------------------------------------
// Edge-parallel aggregation: out[dst] += H[src] * dis[src]*dis[dst]
// 4 features per thread (float4 gather, native f32 atomics into L2).
// ---------------------------------------------------------------------------
__global__ __launch_bounds__(256) void k_aggregate(const long long* __restrict__ ei,
                                                   const float* __restrict__ H,
                                                   const float* __restrict__ dis,
                                                   float* __restrict__ out,
                                                   int E, int Dout) {
    int idx = blockIdx.x * blockDim.x + threadIdx.x;
    int ch  = Dout >> 2;            // feature chunks of 4
    int e   = idx / ch;
    if (e >= E) return;
    int f0 = (idx - e * ch) << 2;

    long long s = ei[e];
    long long d = ei[(long long)E + e];
    float nrm = dis[s] * dis[d];

    const float4 hv = *(const float4*)(H + (size_t)s * Dout + f0);
    float* o = out + (size_t)d * Dout + f0;
    unsafeAtomicAdd(o + 0, hv.x * nrm);
    unsafeAtomicAdd(o + 1, hv.y * nrm);
    unsafeAtomicAdd(o + 2, hv.z * nrm);
    unsafeAtomicAdd(o + 3, hv.w * nrm);
}

// ---------------------------------------------------------------------------
// Finalize: out = [relu](agg + H*dis^2 (self-loop) + bias)
// ---------------------------------------------------------------------------
__global__ __launch_bounds__(256) void k_finalize(const float* __restrict__ agg,
                                                  const float* __restrict__ H,
                                                  const float* __restrict__ dis,
                                                  const float* __restrict__ bias,
                                                  float* __restrict__ out,
                                                  int n, int Dout, int do_relu) {
    int idx = blockIdx.x * blockDim.x + threadIdx.x;
    int total = n * Dout;
    if (idx >= total) return;
    int i = idx / Dout;
    int f = idx - i * Dout;
    float di = dis[i];
    float v = agg[idx] + H[idx] * di * di + bias[f];
    out[idx] = do_relu ? fmaxf(v, 0.0f) : v;
}

// ---------------------------------------------------------------------------
// Host orchestration
// ---------------------------------------------------------------------------
extern "C" void kernel_launch(void* const* d_in, const int* in_sizes, int n_in,
                              void* d_out, int out_size, void* d_ws, size_t ws_size,
                              hipStream_t stream) {
    const float*     x  = (const float*)d_in[0];
    const long long* ei = (const long long*)d_in[1];
    const float* W1 = (const float*)d_in[2]; const float* b1 = (const float*)d_in[3];
    const float* W2 = (const float*)d_in[4]; const float* b2 = (const float*)d_in[5];
    const float* W3 = (const float*)d_in[6]; const float* b3 = (const float*)d_in[7];
    const float* W4 = (const float*)d_in[8]; const float* b4 = (const float*)d_in[9];

    const int D_IN = 128, D_H = 64;
    int N = in_sizes[0] / D_IN;   // 50000
    int E = in_sizes[1] / 2;      // 640000

    // Workspace layout
    float* dis = (float*)d_ws;                 // N
    float* H   = dis + N;                      // N*128 (max GEMM output)
    float* AGG = H + (size_t)N * D_IN;         // N*128 (aggregation accumulator)
    float* TMP = AGG + (size_t)N * D_IN;       // N*64  (intermediate activations)

    // Output layout: x_hat [N,128] then z [N,64]
    float* xhat = (float*)d_out;
    float* z    = xhat + (size_t)N * D_IN;

    const int T = 256;
    // --- degree / normalization ---
    k_init_deg<<<(N + T - 1) / T, T, 0, stream>>>(dis, N);
    k_count_deg<<<(E + T - 1) / T, T, 0, stream>>>(ei, dis, E);
    k_deg_inv_sqrt<<<(N + T - 1) / T, T, 0, stream>>>(dis, N);

    auto layer = [&](const float* in, const float* W, const float* b,
                     float* out, int Din, int Dout, int relu) {
        int tiles  = (N / 16) * (Dout / 16);
        int blocks = (tiles + 7) / 8;  // 8 waves per 256-thread block
        k_gemm_wmma_f32<<<blocks, T, 0, stream>>>(in, W, H, N, Din, Dout);
        hipMemsetAsync(AGG, 0, (size_t)N * Dout * sizeof(float), stream);
        long long work = (long long)E * (Dout / 4);
        k_aggregate<<<(int)((work + T - 1) / T), T, 0, stream>>>(ei, H, dis, AGG, E, Dout);
        int tot = N * Dout;
        k_finalize<<<(tot + T - 1) / T, T, 0, stream>>>(AGG, H, dis, b, out, N, Dout, relu);
    };

    layer(x,   W1, b1, TMP,  D_IN, D_H, 1);  // h  = relu(conv1(x))
    layer(TMP, W2, b2, z,    D_H,  D_H, 1);  // z  = relu(conv2(h))   -> d_out slice
    layer(z,   W3, b3, TMP,  D_H,  D_H, 1);  // x3 = relu(conv3(z))
    layer(TMP, W4, b4, xhat, D_H,  D_IN, 0); // x_hat = conv4(x3)     -> d_out
}